// Quasar1ParameterLNN_37778532336361
// MI455X (gfx1250) — compile-verified
//
#include <hip/hip_runtime.h>
#include <stdint.h>

// ---------------- constants ----------------
#define BB   4
#define SS   512
#define HH   1024
#define LL   4
#define VV   32000
#define BS   (BB*SS)          // 2048
#define SH   (SS*HH)          // 524288
#define HHH  (HH*HH)          // 1048576
#define DT   0.1f
#define HPL  (16*HH)          // padded layer stride (16 rows, rows 4..15 = 0)
#define HSZ  (LL*HPL)         // 65536 floats per h buffer

// workspace layout (float offsets)
#define WS_BAR   0                       // 64 dwords barrier state
#define WS_H0    64
#define WS_H1    (WS_H0 + HSZ)           // 65600
#define WS_X     (WS_H1 + HSZ)           // 131136
#define WS_OUT   (WS_X  + (size_t)BS*HH) // + 2097152

typedef float v2f __attribute__((ext_vector_type(2)));
typedef float v8f __attribute__((ext_vector_type(8)));
typedef unsigned v4u __attribute__((ext_vector_type(4)));
typedef int v4i __attribute__((ext_vector_type(4)));
typedef int v8i __attribute__((ext_vector_type(8)));

// ---------------- helpers ----------------
__device__ __forceinline__ float softplusf(float v) {
  return (v > 20.0f) ? v : log1pf(expf(v));
}

__device__ __forceinline__ v8f wmma_f32_k4(v2f a, v2f b, v8f c) {
  // D = A(16x4 f32) * B(4x16 f32) + C(16x16 f32)
  return __builtin_amdgcn_wmma_f32_16x16x4_f32(false, a, false, b, (short)0, c,
                                               false, false);
}

// device-wide barrier across NWG workgroups (agent-scope, graph-capture safe)
__device__ __forceinline__ void grid_barrier(unsigned* bar, unsigned nwg) {
  __syncthreads();
  if (threadIdx.x == 0) {
    __builtin_amdgcn_fence(__ATOMIC_RELEASE, "agent");
    unsigned* cnt = bar;
    unsigned* gen = bar + 1;
    unsigned g = __hip_atomic_load(gen, __ATOMIC_RELAXED, __HIP_MEMORY_SCOPE_AGENT);
    unsigned arrived = __hip_atomic_fetch_add(cnt, 1u, __ATOMIC_ACQ_REL,
                                              __HIP_MEMORY_SCOPE_AGENT);
    if (arrived == nwg - 1u) {
      __hip_atomic_store(cnt, 0u, __ATOMIC_RELAXED, __HIP_MEMORY_SCOPE_AGENT);
      __hip_atomic_fetch_add(gen, 1u, __ATOMIC_RELEASE, __HIP_MEMORY_SCOPE_AGENT);
    } else {
      while (__hip_atomic_load(gen, __ATOMIC_ACQUIRE, __HIP_MEMORY_SCOPE_AGENT) == g) {
        __builtin_amdgcn_s_sleep(2);
      }
    }
    __builtin_amdgcn_fence(__ATOMIC_ACQUIRE, "agent");
  }
  __syncthreads();
}

// ---------------- kernel 0: init state + barrier ----------------
__global__ __launch_bounds__(256) void k_init(float* ws) {
  unsigned* bar = (unsigned*)ws;
  if (threadIdx.x < 64) bar[threadIdx.x] = 0u;
  float* h0 = ws + WS_H0;
  float* h1 = ws + WS_H1;
  for (int i = threadIdx.x; i < HSZ; i += 256) { h0[i] = 0.0f; h1[i] = 0.0f; }
}

// ---------------- kernel 1: embedding gather ----------------
__global__ __launch_bounds__(256) void k_embed(const int* __restrict__ ids,
                                               const float* __restrict__ emb,
                                               float* __restrict__ x) {
  const int row = blockIdx.x;                    // b*S + s
  const long id = (long)ids[row];
  const float4* src = (const float4*)(emb + id * HH);
  float4* dst = (float4*)(x + (long)row * HH);
  dst[threadIdx.x] = src[threadIdx.x];
}

// ---------------- kernel 2: recurrent scan (persistent, WMMA f32) --------
// 16 WGs x 8 waves = 128 waves. Wave pair (pair,half) owns 16 output columns;
// the two halves split K=1024 and reduce partial C tiles through LDS.
//
// All per-lane address components are hoisted into the six pointers, so the
// K loop has a compile-time trip count (scalar loop control) and the k offset
// folds into load immediates.
template <bool MASKU>
__device__ __forceinline__ void kloop(const float* __restrict__ ph,
                                      const float* __restrict__ pu,
                                      const float* __restrict__ pth,
                                      const float* __restrict__ ptu,
                                      const float* __restrict__ pw,
                                      const float* __restrict__ pup,
                                      float umask, v8f& ct, v8f& ca) {
  #pragma unroll 8
  for (int k = 0; k < HH/2; k += 4) {
    float2 t;
    t = *(const float2*)(ph + k);                 // padded rows 4..15 read 0
    v2f a_h = { t.x, t.y };
    t = *(const float2*)(pu + k);
    v2f a_u;
    if (MASKU) a_u = (v2f){ t.x * umask, t.y * umask };
    else       a_u = (v2f){ t.x, t.y };
    float2 w;
    w = *(const float2*)(pth + k);
    ct = wmma_f32_k4(a_h, (v2f){w.x, w.y}, ct);
    w = *(const float2*)(ptu + k);
    ct = wmma_f32_k4(a_u, (v2f){w.x, w.y}, ct);
    w = *(const float2*)(pw + k);
    ca = wmma_f32_k4(a_h, (v2f){w.x, w.y}, ca);
    w = *(const float2*)(pup + k);
    ca = wmma_f32_k4(a_u, (v2f){w.x, w.y}, ca);
  }
}

__global__ __launch_bounds__(256, 1) void k_scan(
    const float* __restrict__ x,
    const float* __restrict__ Ws,  const float* __restrict__ Us,
    const float* __restrict__ bs,
    const float* __restrict__ twh, const float* __restrict__ twu,
    const float* __restrict__ tb,
    const float* __restrict__ lam,
    float* __restrict__ h0, float* __restrict__ h1,
    float* __restrict__ out_seq, unsigned* bar) {
  const int tid  = threadIdx.x;
  const int lane = tid & 31;
  const int wv   = tid >> 5;            // 0..7
  const int pair = wv >> 1;             // 0..3 : which column tile in this WG
  const int half = wv & 1;              // 0..1 : K-half
  const int n0   = (blockIdx.x * 4 + pair) << 4;   // column base, 0..1008
  const int lm   = lane & 15;
  const int koff = (lane >> 4) << 1;
  const int kbeg = half * (HH/2);
  const int mclp = (lm < 4) ? lm : 3;
  const float um0 = (lm < 4) ? 1.0f : 0.0f;
  const float pos_lambda = log1pf(expf(lam[0]));

  // per-lane invariant element offsets
  const long woff = (long)(n0 + lm) * HH + koff + kbeg;   // into weight matrices
  const long hoff = (long)lm * HH + koff + kbeg;          // into padded h buffers
  const long xoff = (long)mclp * SH + koff + kbeg;        // into x (masked rows)

  __shared__ float red[4][2][4][16];    // pair, {tau,act}, reg, lane  (2 KB)

  // step-invariant biases for this lane's column
  float tbv[LL], bsv[LL];
  #pragma unroll
  for (int i = 0; i < LL; ++i) {
    tbv[i] = tb[i*HH + n0 + lm];
    bsv[i] = bs[i*HH + n0 + lm];
  }
  float hreg[LL][BB];                   // this lane's column of h, all layers
  #pragma unroll
  for (int i = 0; i < LL; ++i)
    #pragma unroll
    for (int r = 0; r < BB; ++r) hreg[i][r] = 0.0f;

  for (int s = 0; s < SS; ++s) {
    const float* hcur = (s & 1) ? h1 : h0;
    float*       hnxt = (s & 1) ? h0 : h1;
    for (int i = 0; i < LL; ++i) {
      v8f ct = {0.f,0.f,0.f,0.f,0.f,0.f,0.f,0.f};
      v8f ca = {0.f,0.f,0.f,0.f,0.f,0.f,0.f,0.f};

      const float* ph  = hcur + (long)i * HPL + hoff;
      const float* pth = twh + (long)i * HHH + woff;
      const float* ptu = twu + (long)i * HHH + woff;
      const float* pw  = Ws  + (long)i * HHH + woff;
      const float* pup = Us  + (long)i * HHH + woff;

      if (i == 0) {
        kloop<true >(ph, x + (long)s*HH + xoff, pth, ptu, pw, pup, um0, ct, ca);
      } else {
        kloop<false>(ph, hnxt + (long)(i-1)*HPL + hoff, pth, ptu, pw, pup,
                     um0, ct, ca);
      }

      // cross-half reduction of the two partial C tiles (only regs 0..3,
      // lanes 0..15 carry the real 4 batch rows)
      if (half == 1 && lane < 16) {
        #pragma unroll
        for (int r = 0; r < BB; ++r) {
          red[pair][0][r][lm] = ct[r];
          red[pair][1][r][lm] = ca[r];
        }
      }
      __syncthreads();
      if (half == 0 && lane < 16) {
        const int n = n0 + lm;
        #pragma unroll
        for (int r = 0; r < BB; ++r) {
          float pt  = ct[r] + red[pair][0][r][lm] + tbv[i];
          float pa  = ca[r] + red[pair][1][r][lm] + bsv[i];
          float tau = softplusf(pt) + 0.01f;
          float act = tanhf(pa);
          float hv  = hreg[i][r];
          float dx  = pos_lambda * (-hv / tau + act);
          dx = fminf(10.0f, fmaxf(-10.0f, dx));
          float hn = hv + DT * dx;
          hreg[i][r] = hn;
          hnxt[(long)i*HPL + (long)r*HH + n] = hn;
          if (i == LL-1) out_seq[(long)r*SH + (long)s*HH + n] = hn;
        }
      }
      grid_barrier(bar, 16u);
    }
  }
}

// ---------------- kernel 3: projection GEMM (pipelined TDM + WMMA) -------
// C[2048 x 32000] = A[2048 x 1024] * proj_w^T + proj_b
// 64x64 C tile / WG, BK=64; TDM double-buffers LDS tiles (rows padded 64->68
// dwords via the TDM pad feature to kill stride-64 bank conflicts).
#define LROW 68

#if __has_builtin(__builtin_amdgcn_tensor_load_to_lds)
#define HAVE_TDM 1
__device__ __forceinline__ void tdm_load_64x64(uint32_t lds_off,
                                               const float* gtile,
                                               uint32_t tensor_rows) {
  const uint64_t ga = (uint64_t)gtile;
  v4u g0 = { 1u,                                   // count=1, user descriptor
             lds_off,                              // LDS byte address
             (uint32_t)ga,                         // global_addr[31:0]
             (uint32_t)((ga >> 32) & 0x1FFFFFFu) | (2u << 30) };  // [56:32]|type=2
  v8i g1 = { (int)((2u<<16) | (1u<<20) | (5u<<22) | (3u<<25)),
             // data_size=4B | pad_enable | pad every 64 dw | pad 4 dw
             (int)((HH & 0xFFFFu) << 16),          // tensor_dim0 lo16
             (int)(((HH >> 16) & 0xFFFFu) | ((tensor_rows & 0xFFFFu) << 16)),
             (int)(((tensor_rows >> 16) & 0xFFFFu) | (64u << 16)), // tile_dim0=64
             (int)64,                              // tile_dim1=64
             (int)HH,                              // tensor_dim0_stride lo32
             0, 0 };
  v4i z4 = {0,0,0,0};
#if defined(__clang_major__) && __clang_major__ >= 23
  v8i z8 = {0,0,0,0,0,0,0,0};
  __builtin_amdgcn_tensor_load_to_lds(g0, g1, z4, z4, z8, 0);
#else
  __builtin_amdgcn_tensor_load_to_lds(g0, g1, z4, z4, 0);
#endif
}
#endif

__device__ __forceinline__ void proj_mac64(const float* __restrict__ La,
                                           const float* __restrict__ Lb,
                                           int widM, int widN, int lm, int koff,
                                           v8f& c0, v8f& c1) {
  const float* pa0 = &La[(widM*32 + lm)      * LROW + koff];
  const float* pa1 = &La[(widM*32 + 16 + lm) * LROW + koff];
  const float* pbb = &Lb[(widN*16 + lm)      * LROW + koff];
  #pragma unroll 8
  for (int kk = 0; kk < 64; kk += 4) {
    float2 b2 = *(const float2*)(pbb + kk);
    v2f bf = { b2.x, b2.y };
    float2 a2 = *(const float2*)(pa0 + kk);
    c0 = wmma_f32_k4((v2f){a2.x, a2.y}, bf, c0);
    a2 = *(const float2*)(pa1 + kk);
    c1 = wmma_f32_k4((v2f){a2.x, a2.y}, bf, c1);
  }
}

__global__ __launch_bounds__(256, 2) void k_proj(const float* __restrict__ A,
                                                 const float* __restrict__ W,
                                                 const float* __restrict__ pb,
                                                 float* __restrict__ out) {
  __shared__ float lsA[2][64 * LROW];
  __shared__ float lsB[2][64 * LROW];
  const int tid  = threadIdx.x;
  const int lane = tid & 31;
  const int wid  = tid >> 5;
  const int widM = wid >> 2;            // 0..1
  const int widN = wid & 3;             // 0..3
  const int lm   = lane & 15;
  const int koff = (lane >> 4) << 1;
  const int m0   = blockIdx.x * 64;
  const int n0g  = blockIdx.y * 64;

  v8f c0 = {0.f,0.f,0.f,0.f,0.f,0.f,0.f,0.f};
  v8f c1 = {0.f,0.f,0.f,0.f,0.f,0.f,0.f,0.f};

#ifdef HAVE_TDM
  if (tid < 32) {                       // wave 0 drives the DMA pipeline
    tdm_load_64x64((uint32_t)(uintptr_t)&lsA[0][0], A + (long)m0*HH, BS);
    tdm_load_64x64((uint32_t)(uintptr_t)&lsB[0][0], W + (long)n0g*HH, VV);
  }
  for (int t = 0; t < 16; ++t) {
    const int cur = t & 1;
    if (tid < 32) {
      if (t < 15) {                     // prefetch next K block into other buf
        const int kn = (t + 1) << 6;
        tdm_load_64x64((uint32_t)(uintptr_t)&lsA[cur ^ 1][0], A + (long)m0*HH + kn, BS);
        tdm_load_64x64((uint32_t)(uintptr_t)&lsB[cur ^ 1][0], W + (long)n0g*HH + kn, VV);
        __builtin_amdgcn_s_wait_tensorcnt(2);   // in-order: current buf done
      } else {
        __builtin_amdgcn_s_wait_tensorcnt(0);
      }
    }
    __syncthreads();                    // publish current buffer
    proj_mac64(&lsA[cur][0], &lsB[cur][0], widM, widN, lm, koff, c0, c1);
    __syncthreads();                    // done reading before buf is re-DMAed
  }
#else
  for (int k0 = 0; k0 < HH; k0 += 64) {
    __syncthreads();
    for (int idx = tid; idx < 64*64; idx += 256) {
      int r = idx >> 6, c = idx & 63;
      lsA[0][r*LROW + c] = A[(long)(m0 + r)*HH + k0 + c];
      lsB[0][r*LROW + c] = W[(long)(n0g + r)*HH + k0 + c];
    }
    __syncthreads();
    proj_mac64(&lsA[0][0], &lsB[0][0], widM, widN, lm, koff, c0, c1);
  }
#endif

  const int n = n0g + widN*16 + lm;
  const float bias = pb[n];
  const int mhi = (lane < 16) ? 0 : 8;
  #pragma unroll
  for (int r = 0; r < 8; ++r) {
    int m = m0 + widM*32 + r + mhi;
    out[(long)m * VV + n]        = c0[r] + bias;
    out[(long)(m + 16) * VV + n] = c1[r] + bias;
  }
}

// ---------------- kernel 4: copy final hidden state hT ----------------
// hT layout [L][B][H]; h buffer layout [L][16][H] (rows 4..15 are padding)
__global__ __launch_bounds__(256) void k_copy_ht(const float* __restrict__ h0,
                                                 float* __restrict__ dst) {
  int i = blockIdx.x * 256 + threadIdx.x;
  if (i < LL*BB*HH) dst[i] = h0[(long)(i >> 12) * HPL + (i & 4095)];
}

// ---------------- launcher ----------------
extern "C" void kernel_launch(void* const* d_in, const int* in_sizes, int n_in,
                              void* d_out, int out_size, void* d_ws, size_t ws_size,
                              hipStream_t stream) {
  (void)in_sizes; (void)n_in; (void)ws_size;
  const int*   ids  = (const int*)  d_in[0];
  const float* emb  = (const float*)d_in[1];
  const float* pw   = (const float*)d_in[2];
  const float* pb   = (const float*)d_in[3];
  const float* lam  = (const float*)d_in[4];
  const float* Ws   = (const float*)d_in[5];
  const float* Us   = (const float*)d_in[6];
  const float* bs   = (const float*)d_in[7];
  const float* twh  = (const float*)d_in[8];
  const float* twu  = (const float*)d_in[9];
  const float* tb   = (const float*)d_in[10];

  float* ws  = (float*)d_ws;
  unsigned* bar = (unsigned*)d_ws;
  float* h0  = ws + WS_H0;
  float* h1  = ws + WS_H1;
  float* x   = ws + WS_X;
  float* oseq= ws + WS_OUT;
  float* logits = (float*)d_out;
  float* hT     = (float*)d_out + (out_size - LL*BB*HH);

  k_init <<<1,  256, 0, stream>>>(ws);
  k_embed<<<BS, 256, 0, stream>>>(ids, emb, x);
  k_scan <<<16, 256, 0, stream>>>(x, Ws, Us, bs, twh, twu, tb, lam,
                                  h0, h1, oseq, bar);
  k_proj <<<dim3(BS/64, VV/64), 256, 0, stream>>>(oseq, pw, pb, logits);
  k_copy_ht<<<(LL*BB*HH + 255)/256, 256, 0, stream>>>(h0, hT);
}